// HGTTransformerLayer_42494406427177
// MI455X (gfx1250) — compile-verified
//
#include <hip/hip_runtime.h>
#include <math.h>

// ---------------------------------------------------------------------------
// HGT transformer layer for MI455X (gfx1250), fp32 throughout.
// Dense GEMMs: V_WMMA_F32_16X16X4_F32, 16x64 output tile per wave32
// (4 accumulators, A-fragment reused 4x, K-loop unrolled 4x -> 16 WMMA/body).
// Edge softmax/aggregation: hoisted per-node head transforms + f32 atomics.
// Workspace: ~N*1936 + 16E floats (~407 MB for N=50000, E=300000).
// ---------------------------------------------------------------------------

typedef __attribute__((ext_vector_type(2))) float v2f;
typedef __attribute__((ext_vector_type(8))) float v8f;

#define WPB 4  // waves per block in WMMA GEMM kernels

__device__ __forceinline__ float gelu_exact(float x) {
  return 0.5f * x * (1.0f + erff(x * 0.70710678118654752f));
}

// Ordered-int float atomic max (any sign; init location to -inf).
__device__ __forceinline__ void atomic_max_float(float* addr, float val) {
  if (val >= 0.0f) {
    atomicMax((int*)addr, __float_as_int(val));
  } else {
    atomicMin((unsigned int*)addr, (unsigned int)__float_as_int(val));
  }
}

__device__ __forceinline__ v8f wmma_f32(v2f a, v2f b, v8f c) {
  return __builtin_amdgcn_wmma_f32_16x16x4_f32(false, a, false, b,
                                               (short)0, c, false, false);
}

// ---------------------------------------------------------------------------
// fp32 WMMA GEMM: C[M,Ncols] = epi(A[M,K] @ W[K,Ncols] + bias)
// One wave computes a 16x64 tile (4 x 16x16 accumulators). Ntiles % 4 == 0.
// EPI: 0 = store, 1 = exact GELU, 2 = + res (residual),
//      3 = skip-gate: C = aux_x + sig*(acc+bias) + (1-sig)*aux_xn
// ---------------------------------------------------------------------------
template <int EPI>
__global__ void wmma_gemm_kernel(const float* __restrict__ A, int lda,
                                 const float* __restrict__ W, int ldb,
                                 const float* __restrict__ bias,
                                 const float* __restrict__ res,
                                 const float* __restrict__ aux_x,
                                 const float* __restrict__ aux_xn,
                                 const float* __restrict__ skip,
                                 float* __restrict__ C, int ldc,
                                 int Mtiles, int Ngroups, int K) {
  const int t = blockIdx.x * WPB + (threadIdx.x >> 5);
  if (t >= Mtiles * Ngroups) return;  // uniform per wave -> EXEC all-1s
  const int tm = t / Ngroups;
  const int tn0 = (t % Ngroups) * 4;  // first of 4 adjacent 16-col tiles
  const int lane = threadIdx.x & 31;
  const int half = lane >> 4;  // 0: K pair {0,1}, 1: K pair {2,3}
  const int m = lane & 15;

  const float* Ap = A + (size_t)(tm * 16 + m) * lda + half * 2;
  const float* Wp = W + (size_t)(half * 2) * ldb + (tn0 * 16 + m);
  __builtin_prefetch(Ap, 0, 3);  // global_prefetch_b8: pull A row into WGP$

  v8f acc[4] = {v8f{}, v8f{}, v8f{}, v8f{}};
#pragma unroll 4
  for (int k0 = 0; k0 < K; k0 += 4) {
    float2 av = *(const float2*)(Ap + k0);
    v2f a; a.x = av.x; a.y = av.y;
    const float* wk = Wp + (size_t)k0 * ldb;
#pragma unroll
    for (int j = 0; j < 4; ++j) {
      v2f b; b.x = wk[j * 16]; b.y = wk[j * 16 + ldb];
      acc[j] = wmma_f32(a, b, acc[j]);
    }
  }

  float sig = 0.0f;
  if (EPI == 3) sig = 1.0f / (1.0f + expf(-skip[0]));
#pragma unroll
  for (int j = 0; j < 4; ++j) {
    const int col = (tn0 + j) * 16 + m;
    const float bv = bias[col];
#pragma unroll
    for (int r = 0; r < 8; ++r) {
      const size_t idx = (size_t)(tm * 16 + half * 8 + r) * ldc + col;
      float v = acc[j][r] + bv;
      if (EPI == 1) v = gelu_exact(v);
      if (EPI == 2) v += res[idx];
      if (EPI == 3) v = aux_x[idx] + sig * v + (1.0f - sig) * aux_xn[idx];
      C[idx] = v;
    }
  }
}

// ---------------------------------------------------------------------------
// Per-head relation transforms: kf = k@a_f, vf = v@m_f, kg = k@a_g, vg = v@m_g
// grid = (N/16, H), block = 128 (wave w picks one of the 4 output matrices).
// kqv layout [N,384]: k = cols 0..127, q = 128..255, v = 256..383.
// ---------------------------------------------------------------------------
__global__ void head_transform_kernel(const float* __restrict__ kqv,
                                      const float* __restrict__ a_f,
                                      const float* __restrict__ m_f,
                                      const float* __restrict__ a_g,
                                      const float* __restrict__ m_g,
                                      float* __restrict__ kf, float* __restrict__ vf,
                                      float* __restrict__ kg, float* __restrict__ vg) {
  const int w = threadIdx.x >> 5;  // 0..3
  const int lane = threadIdx.x & 31;
  const int half = lane >> 4;
  const int m = lane & 15;
  const int tm = blockIdx.x;
  const int h = blockIdx.y;

  const float* A = kqv + ((w & 1) ? 256 : 0) + h * 16;  // k (w even) / v (w odd)
  const float* B = (w == 0 ? a_f : w == 1 ? m_f : w == 2 ? a_g : m_g) + h * 256;
  float* Cb = (w == 0 ? kf : w == 1 ? vf : w == 2 ? kg : vg) + h * 16;

  const float* Ap = A + (size_t)(tm * 16 + m) * 384 + half * 2;
  const float* Wp = B + (half * 2) * 16 + m;

  v8f acc = {};
#pragma unroll
  for (int k0 = 0; k0 < 16; k0 += 4) {
    float2 av = *(const float2*)(Ap + k0);
    v2f a; a.x = av.x; a.y = av.y;
    v2f b; b.x = Wp[k0 * 16]; b.y = Wp[k0 * 16 + 16];
    acc = wmma_f32(a, b, acc);
  }
#pragma unroll
  for (int r = 0; r < 8; ++r)
    Cb[(size_t)(tm * 16 + half * 8 + r) * 128 + m] = acc[r];
}

// ---------------------------------------------------------------------------
// LayerNorm: one wave32 per row of 128; lane holds 4 strided elements.
// ---------------------------------------------------------------------------
__global__ void ln_kernel(const float* __restrict__ x, const float* __restrict__ g,
                          const float* __restrict__ b, float* __restrict__ y, int N) {
  const int wid = threadIdx.x >> 5;
  const int lane = threadIdx.x & 31;
  const int row = blockIdx.x * (blockDim.x >> 5) + wid;
  if (row >= N) return;
  const float* xr = x + (size_t)row * 128;
  float v[4];
  float s = 0.0f;
#pragma unroll
  for (int j = 0; j < 4; ++j) { v[j] = xr[lane + 32 * j]; s += v[j]; }
#pragma unroll
  for (int o = 16; o > 0; o >>= 1) s += __shfl_xor(s, o, 32);
  const float mu = s * (1.0f / 128.0f);
  float vs = 0.0f;
#pragma unroll
  for (int j = 0; j < 4; ++j) { float d = v[j] - mu; vs += d * d; }
#pragma unroll
  for (int o = 16; o > 0; o >>= 1) vs += __shfl_xor(vs, o, 32);
  const float rstd = rsqrtf(vs * (1.0f / 128.0f) + 1e-5f);
  float* yr = y + (size_t)row * 128;
#pragma unroll
  for (int j = 0; j < 4; ++j) {
    const int c = lane + 32 * j;
    yr[c] = (v[j] - mu) * rstd * g[c] + b[c];
  }
}

// ---------------------------------------------------------------------------
// Edge pipeline (idx over 2E*8 = (edge, head)); type-0 edges first.
// ---------------------------------------------------------------------------
__global__ void init_kernel(float* __restrict__ mx, float* __restrict__ den,
                            float* __restrict__ agg, int n8, int n128) {
  const int i = blockIdx.x * blockDim.x + threadIdx.x;
  if (i < n128) agg[i] = 0.0f;
  if (i < n8) { mx[i] = -INFINITY; den[i] = 0.0f; }
}

__global__ void edge_pass1(const int* __restrict__ ef, const int* __restrict__ eg,
                           const float* __restrict__ kqv,
                           const float* __restrict__ kf, const float* __restrict__ kg,
                           const float* __restrict__ p_f, const float* __restrict__ p_g,
                           float* __restrict__ loge, float* __restrict__ mx, int E) {
  const int idx = blockIdx.x * blockDim.x + threadIdx.x;
  if (idx >= 2 * E * 8) return;
  const int h = idx & 7;
  const int e = idx >> 3;
  int src, dst; const float* kbuf; float p;
  if (e < E) { src = ef[e]; dst = ef[E + e]; kbuf = kf; p = p_f[h]; }
  else       { src = eg[e - E]; dst = eg[E + (e - E)]; kbuf = kg; p = p_g[h]; }
  const float4* qp = (const float4*)(kqv + (size_t)dst * 384 + 128 + h * 16);
  const float4* kp = (const float4*)(kbuf + (size_t)src * 128 + h * 16);
  float d = 0.0f;
#pragma unroll
  for (int j = 0; j < 4; ++j) {
    float4 a = qp[j], b = kp[j];
    d += a.x * b.x + a.y * b.y + a.z * b.z + a.w * b.w;
  }
  const float logit = d * p * 0.25f;  // 1/sqrt(DH) = 0.25
  loge[idx] = logit;
  atomic_max_float(&mx[dst * 8 + h], logit);
}

__global__ void edge_pass2(const int* __restrict__ ef, const int* __restrict__ eg,
                           float* __restrict__ loge, const float* __restrict__ mx,
                           float* __restrict__ den, int E) {
  const int idx = blockIdx.x * blockDim.x + threadIdx.x;
  if (idx >= 2 * E * 8) return;
  const int h = idx & 7;
  const int e = idx >> 3;
  const int dst = (e < E) ? ef[E + e] : eg[E + (e - E)];
  const float ex = expf(loge[idx] - mx[dst * 8 + h]);  // mx finite for any used dst
  loge[idx] = ex;
  atomicAdd(&den[dst * 8 + h], ex);
}

__global__ void edge_pass3(const int* __restrict__ ef, const int* __restrict__ eg,
                           const float* __restrict__ loge,
                           const float* __restrict__ vf, const float* __restrict__ vg,
                           float* __restrict__ agg, int E) {
  const int idx = blockIdx.x * blockDim.x + threadIdx.x;
  if (idx >= 2 * E * 8) return;
  const int h = idx & 7;
  const int e = idx >> 3;
  int src, dst; const float* vbuf;
  if (e < E) { src = ef[e]; dst = ef[E + e]; vbuf = vf; }
  else       { src = eg[e - E]; dst = eg[E + (e - E)]; vbuf = vg; }
  const float ex = loge[idx];
  const float* vp = vbuf + (size_t)src * 128 + h * 16;
  float* ap = agg + (size_t)dst * 128 + h * 16;
#pragma unroll
  for (int j = 0; j < 16; ++j) atomicAdd(&ap[j], ex * vp[j]);
}

__global__ void finalize_kernel(float* __restrict__ agg, const float* __restrict__ den,
                                int n128) {
  const int idx = blockIdx.x * blockDim.x + threadIdx.x;
  if (idx >= n128) return;
  const int n = idx >> 7;
  const int h = (idx & 127) >> 4;
  const float d = den[n * 8 + h] + 1e-16f;
  agg[idx] = gelu_exact(agg[idx] / d);
}

// ---------------------------------------------------------------------------
static inline int cdiv(long a, long b) { return (int)((a + b - 1) / b); }

template <int EPI>
static void launch_gemm(const float* A, int lda, const float* W, int ldb,
                        const float* bias, const float* res, const float* aux_x,
                        const float* aux_xn, const float* skip, float* C, int ldc,
                        int Mt, int Ngroups, int K, hipStream_t s) {
  const int tiles = Mt * Ngroups;
  wmma_gemm_kernel<EPI><<<cdiv(tiles, WPB), WPB * 32, 0, s>>>(
      A, lda, W, ldb, bias, res, aux_x, aux_xn, skip, C, ldc, Mt, Ngroups, K);
}

extern "C" void kernel_launch(void* const* d_in, const int* in_sizes, int n_in,
                              void* d_out, int out_size, void* d_ws, size_t ws_size,
                              hipStream_t stream) {
  const float* x     = (const float*)d_in[0];
  const int*   ef    = (const int*)d_in[1];
  const int*   eg    = (const int*)d_in[2];
  const float* kqv_w = (const float*)d_in[3];
  const float* kqv_b = (const float*)d_in[4];
  const float* a_f   = (const float*)d_in[5];
  const float* m_f   = (const float*)d_in[6];
  const float* p_f   = (const float*)d_in[7];
  const float* a_g   = (const float*)d_in[8];
  const float* m_g   = (const float*)d_in[9];
  const float* p_g   = (const float*)d_in[10];
  const float* out_w = (const float*)d_in[11];
  const float* out_b = (const float*)d_in[12];
  const float* skip  = (const float*)d_in[13];
  const float* ln1_g = (const float*)d_in[14];
  const float* ln1_b = (const float*)d_in[15];
  const float* ln2_g = (const float*)d_in[16];
  const float* ln2_b = (const float*)d_in[17];
  const float* w1    = (const float*)d_in[18];
  const float* b1    = (const float*)d_in[19];
  const float* w2    = (const float*)d_in[20];
  const float* b2    = (const float*)d_in[21];

  const int N = in_sizes[0] / 128;  // 50000 (multiple of 16)
  const int E = in_sizes[1] / 2;    // 300000

  // ---- workspace layout (floats) ----
  float* ws = (float*)d_ws;
  size_t off = 0;
  float* xn   = ws + off; off += (size_t)N * 128;
  float* kqv  = ws + off; off += (size_t)N * 384;
  float* kf   = ws + off; off += (size_t)N * 128;
  float* vf   = ws + off; off += (size_t)N * 128;
  float* kg   = ws + off; off += (size_t)N * 128;
  float* vg   = ws + off; off += (size_t)N * 128;
  float* loge = ws + off; off += (size_t)2 * E * 8;
  float* mx   = ws + off; off += (size_t)N * 8;
  float* den  = ws + off; off += (size_t)N * 8;
  float* agg  = ws + off; off += (size_t)N * 128;
  float* x1   = ws + off; off += (size_t)N * 128;
  float* xn2  = ws + off; off += (size_t)N * 128;
  float* h1   = ws + off; off += (size_t)N * 512;
  (void)ws_size; (void)n_in; (void)out_size;

  const int Mt = N / 16;
  const int nEH = 2 * E * 8;

  // 1. init segment-softmax accumulators
  init_kernel<<<cdiv((long)N * 128, 256), 256, 0, stream>>>(mx, den, agg, N * 8, N * 128);
  // 2. LN1
  ln_kernel<<<cdiv(N, 8), 256, 0, stream>>>(x, ln1_g, ln1_b, xn, N);
  // 3. kqv = xn @ kqv_w + kqv_b   [N,384]  (24 col-tiles = 6 groups)
  launch_gemm<0>(xn, 128, kqv_w, 384, kqv_b, nullptr, nullptr, nullptr, nullptr,
                 kqv, 384, Mt, 6, 128, stream);
  // 4. per-head relation transforms
  {
    dim3 g(Mt, 8);
    head_transform_kernel<<<g, 128, 0, stream>>>(kqv, a_f, m_f, a_g, m_g, kf, vf, kg, vg);
  }
  // 5-7. edge softmax + aggregation
  edge_pass1<<<cdiv(nEH, 256), 256, 0, stream>>>(ef, eg, kqv, kf, kg, p_f, p_g, loge, mx, E);
  edge_pass2<<<cdiv(nEH, 256), 256, 0, stream>>>(ef, eg, loge, mx, den, E);
  edge_pass3<<<cdiv(nEH, 256), 256, 0, stream>>>(ef, eg, loge, vf, vg, agg, E);
  // 8. agg -> gelu(agg/denom) in place
  finalize_kernel<<<cdiv((long)N * 128, 256), 256, 0, stream>>>(agg, den, N * 128);
  // 9. out projection with fused skip-gate + residual -> x1
  //    x1 = x + sig*(agg@out_w + out_b) + (1-sig)*xn
  launch_gemm<3>(agg, 128, out_w, 128, out_b, nullptr, x, xn, skip,
                 x1, 128, Mt, 2, 128, stream);
  // 10. LN2
  ln_kernel<<<cdiv(N, 8), 256, 0, stream>>>(x1, ln2_g, ln2_b, xn2, N);
  // 11. FFN up (fused exact GELU)   [N,512]  (32 col-tiles = 8 groups)
  launch_gemm<1>(xn2, 128, w1, 512, b1, nullptr, nullptr, nullptr, nullptr,
                 h1, 512, Mt, 8, 128, stream);
  // 12. FFN down (fused residual add) -> d_out
  launch_gemm<2>(h1, 512, w2, 128, b2, x1, nullptr, nullptr, nullptr,
                 (float*)d_out, 128, Mt, 2, 512, stream);
}